// DynamicAttention_65412351918715
// MI455X (gfx1250) — compile-verified
//
#include <hip/hip_runtime.h>
#include <stdint.h>

#define BB 128
#define SS 2048
#define HH 512
#define SCHUNK 16
#define NCHUNK (SS / SCHUNK)   // 128
#define DEPTH 4                // LDS pipeline depth
#define LROW 516               // padded tile row stride (floats) -> conflict-free banks
#define NTHREADS 256
#define NWAVES 8

typedef float v2f __attribute__((ext_vector_type(2)));
typedef float v8f __attribute__((ext_vector_type(8)));

__device__ __forceinline__ float fast_tanh(float x) {
#if __has_builtin(__builtin_amdgcn_tanhf)
  return __builtin_amdgcn_tanhf(x);
#else
  // tanh(x) = 1 - 2/(exp2(2x*log2e)+1); saturates correctly at +/-1
  float t = __builtin_amdgcn_exp2f(x * 2.885390081777927f);
  return 1.0f - 2.0f * __builtin_amdgcn_rcpf(t + 1.0f);
#endif
}

__device__ __forceinline__ float fast_exp(float x) {
  return __builtin_amdgcn_exp2f(x * 1.4426950408889634f);
}

// CDNA5 async copy: global -> LDS, 16B per lane, tracked by ASYNCcnt
__device__ __forceinline__ void async_b128(uint32_t lds_byte_addr, const float* gaddr) {
  asm volatile("global_load_async_to_lds_b128 %0, %1, off"
               :: "v"(lds_byte_addr), "v"(gaddr)
               : "memory");
}

// One wave issues 8 async b128 ops = 4 KiB = 2 tile rows; 8 waves cover 16 rows.
__device__ __forceinline__ void issue_chunk(const float* lstm_b, int chunk,
                                            uint32_t ltile_base, int w, int lane) {
  const float* g0 = lstm_b + (size_t)chunk * SCHUNK * HH;
#pragma unroll
  for (int j = 0; j < 8; ++j) {
    int row = 2 * w + (j >> 2);
    int col = (j & 3) * 128 + lane * 4;
    const float* g = g0 + row * HH + col;
    uint32_t l = ltile_base + (uint32_t)(row * LROW + col) * 4u;
    async_b128(l, g);
  }
}

__global__ void __launch_bounds__(NTHREADS, 1)
dyn_attn_kernel(const float* __restrict__ lstm,
                const float* __restrict__ vol,
                const float* __restrict__ w_vol,
                const float* __restrict__ w_att,
                float* __restrict__ ctx_out,
                float* __restrict__ attn_out) {
  extern __shared__ __align__(16) float smem[];
  float* Lt    = smem;                          // DEPTH * 16 * LROW
  float* score = Lt + DEPTH * SCHUNK * LROW;    // SS raw scores
  float* vvol  = score + SS;                    // SS
  float* wv    = vvol + SS;                     // HH
  float* wa    = wv + HH;                       // HH
  float* red   = wa + HH;                       // NWAVES*16 per-wave score partials
  float* psm   = red + NWAVES * SCHUNK;         // 16 chunk probabilities
  float* misc  = psm + SCHUNK;                  // [0]=scale [1]=m_final [2]=l_final

  const int tid  = threadIdx.x;
  const int w    = tid >> 5;
  const int lane = tid & 31;
  const int b    = blockIdx.x;

  const float* lstm_b = lstm + (size_t)b * SS * HH;
  const uint32_t Lt_base = (uint32_t)(uintptr_t)Lt;  // low 32 bits of LDS generic addr = LDS offset

  // Prologue: start DEPTH-1 tiles streaming before anything else.
#pragma unroll
  for (int c = 0; c < DEPTH - 1; ++c)
    issue_chunk(lstm_b, c, Lt_base + (uint32_t)(c * SCHUNK * LROW) * 4u, w, lane);

  // Preload volatility row and weight vectors into LDS.
#pragma unroll
  for (int j = 0; j < SS / NTHREADS; ++j)
    vvol[j * NTHREADS + tid] = vol[(size_t)b * SS + j * NTHREADS + tid];
#pragma unroll
  for (int j = 0; j < HH / NTHREADS; ++j) {
    wv[j * NTHREADS + tid] = w_vol[j * NTHREADS + tid];
    wa[j * NTHREADS + tid] = w_att[j * NTHREADS + tid];
  }
  __syncthreads();

  float c0 = 0.f, c1 = 0.f;            // context accumulators: h = tid, tid+256
  float m_run = -__builtin_inff();     // maintained by wave 0
  float l_run = 0.f;

  const int m     = lane & 15;         // A-matrix row (s within chunk)
  const int koff  = (lane >> 4) << 1;  // lanes 0-15: K=0,1 ; lanes 16-31: K=2,3
  const int hbase = w * (HH / NWAVES); // 64 h per wave

  for (int ci = 0; ci < NCHUNK; ++ci) {
    float* buf = Lt + (ci & (DEPTH - 1)) * SCHUNK * LROW;

    if (ci + DEPTH - 1 < NCHUNK)
      issue_chunk(lstm_b, ci + DEPTH - 1,
                  Lt_base + (uint32_t)(((ci + DEPTH - 1) & (DEPTH - 1)) * SCHUNK * LROW) * 4u,
                  w, lane);

    // Drain only the oldest tile (8 async ops per wave per tile, in-order completion).
    int rem = NCHUNK - 1 - ci;
    if (rem >= 3)      asm volatile("s_wait_asynccnt 24" ::: "memory");
    else if (rem == 2) asm volatile("s_wait_asynccnt 16" ::: "memory");
    else if (rem == 1) asm volatile("s_wait_asynccnt 8"  ::: "memory");
    else               asm volatile("s_wait_asynccnt 0"  ::: "memory");
    __syncthreads();  // B1: tile ci visible to all waves

    // ---- Phase 2: scores via f32 WMMA. D[m,n] = sum_k A[m,k]*B[k,n]
    //      A[m,k] = L[s_m,h_k]*tanh(vol[s_m]*w_vol[h_k]);  B[k,n] = w_att[h_k] (bcast over n)
    float vs = vvol[ci * SCHUNK + m];
    v8f acc = {0.f, 0.f, 0.f, 0.f, 0.f, 0.f, 0.f, 0.f};
#pragma unroll
    for (int kk = 0; kk < (HH / NWAVES) / 4; ++kk) {   // 16 chained WMMAs over this wave's h-range
      int h = hbase + kk * 4 + koff;
      float l0 = buf[m * LROW + h];
      float l1 = buf[m * LROW + h + 1];
      v2f a, bm;
      a.x = l0 * fast_tanh(vs * wv[h]);
      a.y = l1 * fast_tanh(vs * wv[h + 1]);
      bm.x = wa[h];
      bm.y = wa[h + 1];
      acc = __builtin_amdgcn_wmma_f32_16x16x4_f32(false, a, false, bm, (short)0, acc,
                                                  false, false);
    }
    // D layout: lane0 holds (m=0..7, n=0) in v0..v7; lane16 holds m=8..15.
    if (lane == 0) {
#pragma unroll
      for (int r = 0; r < 8; ++r) red[w * 16 + r] = acc[r];
    } else if (lane == 16) {
#pragma unroll
      for (int r = 0; r < 8; ++r) red[w * 16 + 8 + r] = acc[r];
    }
    __syncthreads();  // B2

    // ---- Phase 3: wave 0 does online-softmax bookkeeping
    if (w == 0) {
      float sm;
      if (lane < 16) {
        sm = 0.f;
#pragma unroll
        for (int ww = 0; ww < NWAVES; ++ww) sm += red[ww * 16 + lane];
        score[ci * SCHUNK + lane] = sm;          // raw score, normalized at the end
      } else {
        sm = -__builtin_inff();
      }
      float cmax = sm;
#pragma unroll
      for (int off = 16; off >= 1; off >>= 1)
        cmax = fmaxf(cmax, __shfl_xor(cmax, off));
      float m_new = fmaxf(m_run, cmax);
      float scale = fast_exp(m_run - m_new);     // first chunk: exp(-inf)=0
      float p = (lane < 16) ? fast_exp(sm - m_new) : 0.f;
      float sp = p;
#pragma unroll
      for (int off = 16; off >= 1; off >>= 1)
        sp += __shfl_xor(sp, off);
      l_run = l_run * scale + sp;
      m_run = m_new;
      if (lane < 16) psm[lane] = p;
      if (lane == 0) misc[0] = scale;
    }
    __syncthreads();  // B3

    // ---- Phase 4: rescale + accumulate context from the SAME LDS tile (single HBM pass)
    {
      float scale = misc[0];
      c0 *= scale;
      c1 *= scale;
#pragma unroll
      for (int j = 0; j < SCHUNK; ++j) {
        float p = psm[j];
        c0 = fmaf(p, buf[j * LROW + tid], c0);
        c1 = fmaf(p, buf[j * LROW + 256 + tid], c1);
      }
    }
    __syncthreads();  // B4: tile buffer + red/psm free for reuse
  }

  if (tid == 0) { misc[1] = m_run; misc[2] = l_run; }
  __syncthreads();
  float mf  = misc[1];
  float inv = 1.0f / misc[2];

  ctx_out[(size_t)b * HH + tid]       = c0 * inv;
  ctx_out[(size_t)b * HH + 256 + tid] = c1 * inv;
#pragma unroll
  for (int j = 0; j < SS / NTHREADS; ++j) {
    int s = j * NTHREADS + tid;
    attn_out[(size_t)b * SS + s] = fast_exp(score[s] - mf) * inv;
  }
}

extern "C" void kernel_launch(void* const* d_in, const int* in_sizes, int n_in,
                              void* d_out, int out_size, void* d_ws, size_t ws_size,
                              hipStream_t stream) {
  const float* lstm  = (const float*)d_in[0];  // [B,S,H]
  const float* vol   = (const float*)d_in[1];  // [B,S]
  const float* w_vol = (const float*)d_in[2];  // [H]
  const float* w_att = (const float*)d_in[3];  // [H]
  float* out      = (float*)d_out;
  float* ctx_out  = out;            // [B,H]  (first output)
  float* attn_out = out + BB * HH;  // [B,S]  (second output)

  size_t shmem = (size_t)(DEPTH * SCHUNK * LROW + SS + SS + HH + HH +
                          NWAVES * SCHUNK + SCHUNK + 4) * sizeof(float);
  (void)hipFuncSetAttribute((const void*)dyn_attn_kernel,
                            hipFuncAttributeMaxDynamicSharedMemorySize, (int)shmem);

  dyn_attn_kernel<<<BB, NTHREADS, shmem, stream>>>(lstm, vol, w_vol, w_att,
                                                   ctx_out, attn_out);
}